// IPM_29240137351878
// MI455X (gfx1250) — compile-verified
//
#include <hip/hip_runtime.h>
#include <hip/hip_bf16.h>

typedef float v2f __attribute__((ext_vector_type(2)));
typedef float v8f __attribute__((ext_vector_type(8)));

#define W_BEV   200
#define H_BEV   200
#define NPTS    (W_BEV * H_BEV)        // 40000 BEV pixels
#define IMG_H   128
#define IMG_W   352
#define IMG_C   64
#define IMG_HW  (IMG_H * IMG_W)
#define NCAM    6
#define NB      4
#define NK      (NB * NCAM)            // 24 camera views
#define NPIX    (NK * NPTS)            // 960000 projected points
#define GEPS    1e-7f
#define GSTEP   (100.0f / 199.0f)      // linspace(-50,50,200) step

// workspace layout (in floats)
#define WS_M    0                      // 24 x 16 combined matrices M = (Ks@RTs)@R
#define WS_Z    (NK * 16)              // 24 z values
#define WS_PIXO 512                    // pixX[NPIX], then pixY[NPIX]

// ---------------------------------------------------------------------------
// Stage 1: per-view combined matrix M[k] = (Ks[k] @ RTs[k]) @ R[k % 4].
// Replicates the reference's jnp.tile quirk: planes (R, z) use batch k%4,
// while P uses view index k directly.
// ---------------------------------------------------------------------------
__global__ void ipm_setup(const float* __restrict__ Ks,
                          const float* __restrict__ RTs,
                          const float* __restrict__ transl,
                          const float* __restrict__ yrp,
                          float* __restrict__ ws) {
  int k = threadIdx.x;
  if (k >= NK) return;
  int b = k % NB;

  const float* Km = Ks  + k * 16;
  const float* Rt = RTs + k * 16;
  float P[16];
  for (int r = 0; r < 4; ++r)
    for (int c = 0; c < 4; ++c) {
      float s = 0.f;
      for (int t = 0; t < 4; ++t) s += Km[r * 4 + t] * Rt[t * 4 + c];
      P[r * 4 + c] = s;
    }

  // reference: _rotation_from_euler(pitchs, rolls, 0) with
  //   pitchs = yrp[:,2] (-> i angle), rolls = yrp[:,1] (-> j angle), yaw = 0
  const float d2r = 0.017453292519943295f;
  float ai = yrp[b * 3 + 2] * d2r;
  float aj = yrp[b * 3 + 1] * d2r;
  float si = sinf(ai), ci = cosf(ai);
  float sj = sinf(aj), cj = cosf(aj);
  // with sk=0, ck=1:
  float R[16] = { cj,  sj * si, sj * ci, 0.f,
                  0.f, ci,     -si,      0.f,
                 -sj,  cj * si, cj * ci, 0.f,
                  0.f, 0.f,     0.f,     1.f };

  float* M = ws + WS_M + k * 16;
  for (int r = 0; r < 4; ++r)
    for (int c = 0; c < 4; ++c) {
      float s = 0.f;
      for (int t = 0; t < 4; ++t) s += P[r * 4 + t] * R[t * 4 + c];
      M[r * 4 + c] = s;
    }
  ws[WS_Z + k] = transl[b * 3 + 2];
}

// ---------------------------------------------------------------------------
// Stage 2: WMMA projection. One wave projects 16 BEV points through M[k]
// with a single V_WMMA_F32_16X16X4_F32 (K=4 == homogeneous dim).
//   A (16x4): rows 0..2 = M rows 0..2 (rest zero)
//   B (4x16): column n = (X_n, Y_n, z, 1)
//   D (16x16): lane n (<16) holds q0,q1,q2 of point n in d[0],d[1],d[2]
// EXEC is all-ones at the WMMA (only stores are predicated).
// ---------------------------------------------------------------------------
__global__ void ipm_project(const float* __restrict__ ws,
                            float* __restrict__ pixX,
                            float* __restrict__ pixY) {
  int wave = threadIdx.x >> 5;
  int lane = threadIdx.x & 31;
  int k    = blockIdx.y;
  int chunk = blockIdx.x * 8 + wave;           // 16 points per chunk
  if (chunk * 16 >= NPTS) return;              // wave-uniform exit

  const float* M = ws + WS_M + k * 16;
  float z = ws[WS_Z + k];

  int m  = lane & 15;
  int hi = lane >> 4;                          // 0 -> K{0,1}, 1 -> K{2,3}
  // A operand: 16x4, VGPR0 = K0/K2, VGPR1 = K1/K3 (lanes 0-15 / 16-31)
  v2f a;
  a.x = (m < 3) ? M[m * 4 + (hi ? 2 : 0)] : 0.f;
  a.y = (m < 3) ? M[m * 4 + (hi ? 3 : 1)] : 0.f;

  // B operand: 4x16, lane n -> column n (point n), same K striping as A
  int p = chunk * 16 + (lane & 15);
  int i = p / W_BEV;
  int j = p - i * W_BEV;
  float X = -50.f + GSTEP * (float)j;          // coords[0] = y_lin[j]
  float Y = -50.f + GSTEP * (float)i;          // coords[1] = x_lin[i]
  v2f bb;
  bb.x = hi ? z   : X;
  bb.y = hi ? 1.f : Y;

  v8f c = {};
  v8f d = __builtin_amdgcn_wmma_f32_16x16x4_f32(false, a, false, bb,
                                                (short)0, c, false, false);
  if (lane < 16) {
    float q0 = d[0], q1 = d[1], q2 = d[2];
    float inv = 1.0f / (q2 + GEPS);
    pixX[k * NPTS + p] = q0 * inv;
    pixY[k * NPTS + p] = q1 * inv;
  }
}

// ---------------------------------------------------------------------------
// Stage 3: bilinear sampling + mask composition, fused (no 246 MB warped
// tensor). One thread per (b,i,j), channel loop inside; bilinear indices /
// weights hoisted out of the channel loop; where(top==0) fallbacks lazy.
// ---------------------------------------------------------------------------
struct Bil { int i00, i01, i10, i11; float w00, w01, w10, w11; };

__device__ __forceinline__ Bil mkbil(const float* __restrict__ pixX,
                                     const float* __restrict__ pixY,
                                     int k, int p) {
  float px = pixX[k * NPTS + p];
  float py = pixY[k * NPTS + p];
  float x0 = floorf(px), x1 = x0 + 1.f;
  float y0 = floorf(py), y1 = y0 + 1.f;
  const float xm = (float)(IMG_W - 1), ym = (float)(IMG_H - 1);
  x0 = fminf(fmaxf(x0, 0.f), xm);  x1 = fminf(fmaxf(x1, 0.f), xm);
  y0 = fminf(fmaxf(y0, 0.f), ym);  y1 = fminf(fmaxf(y1, 0.f), ym);
  float wx0 = x1 - px, wx1 = px - x0;
  float wy0 = y1 - py, wy1 = py - y0;
  int ix0 = (int)x0, ix1 = (int)x1, iy0 = (int)y0, iy1 = (int)y1;
  Bil r;
  r.i00 = iy0 * IMG_W + ix0;  r.i01 = iy1 * IMG_W + ix0;
  r.i10 = iy0 * IMG_W + ix1;  r.i11 = iy1 * IMG_W + ix1;
  r.w00 = wx0 * wy0;  r.w01 = wx0 * wy1;
  r.w10 = wx1 * wy0;  r.w11 = wx1 * wy1;
  return r;
}

__device__ __forceinline__ float sampl(const float* __restrict__ base, const Bil& s) {
  return s.w00 * base[s.i00] + s.w01 * base[s.i01] +
         s.w10 * base[s.i10] + s.w11 * base[s.i11];
}

__global__ void ipm_compose(const float* __restrict__ images,
                            const float* __restrict__ pixX,
                            const float* __restrict__ pixY,
                            float* __restrict__ out) {
  int t = blockIdx.x * blockDim.x + threadIdx.x;
  if (t >= NB * NPTS) return;
  int j = t % W_BEV;
  int i = (t / W_BEV) % H_BEV;
  int b = t / NPTS;
  int p = i * W_BEV + j;

  // masks (tri[i,j] = i>=j, ftri[i,j] = i>=199-j, front j>=100)
  int camP = (j >= W_BEV / 2) ? 1 : 4;         // CAM_F : CAM_B
  bool fl = (i >= (H_BEV - 1) - j);            // ftri
  bool fr = (i <  j);                          // 1 - tri
  bool bl = (i >= j);                          // tri
  bool br = (i <  (H_BEV - 1) - j);            // 1 - ftri

  int kP  = b * NCAM + camP;
  int kFL = b * NCAM + 0, kFR = b * NCAM + 2;
  int kBL = b * NCAM + 3, kBR = b * NCAM + 5;

  Bil sP  = mkbil(pixX, pixY, kP,  p);
  Bil sFL = mkbil(pixX, pixY, kFL, p);
  Bil sFR = mkbil(pixX, pixY, kFR, p);
  Bil sBL = mkbil(pixX, pixY, kBL, p);
  Bil sBR = mkbil(pixX, pixY, kBR, p);

  const size_t HW = (size_t)IMG_HW;
  const float* basP = images + (size_t)kP * IMG_C * HW;

  for (int c = 0; c < IMG_C; ++c) {
    const float* bp = basP + (size_t)c * HW;
    // prefetch next channel's primary texels (global_prefetch_b8)
    if (c + 1 < IMG_C) {
      __builtin_prefetch(bp + HW + sP.i00, 0, 0);
      __builtin_prefetch(bp + HW + sP.i11, 0, 0);
    }
    float v = sampl(bp, sP);
    if (v == 0.0f) {                           // where(top==0, wfl+wfr, top)
      float v1 = 0.f;
      if (fl) v1 += sampl(images + ((size_t)kFL * IMG_C + c) * HW, sFL);
      if (fr) v1 += sampl(images + ((size_t)kFR * IMG_C + c) * HW, sFR);
      v = v1;
      if (v == 0.0f) {                         // where(==0, wbl+wbr, .)
        float v2 = 0.f;
        if (bl) v2 += sampl(images + ((size_t)kBL * IMG_C + c) * HW, sBL);
        if (br) v2 += sampl(images + ((size_t)kBR * IMG_C + c) * HW, sBR);
        v = v2;
      }
    }
    out[((size_t)(b * IMG_C + c) * H_BEV + i) * W_BEV + j] = v;
  }
}

// ---------------------------------------------------------------------------
extern "C" void kernel_launch(void* const* d_in, const int* in_sizes, int n_in,
                              void* d_out, int out_size, void* d_ws, size_t ws_size,
                              hipStream_t stream) {
  const float* images = (const float*)d_in[0];   // (4,6,64,128,352) f32
  const float* Ks     = (const float*)d_in[1];   // (24,4,4)
  const float* RTs    = (const float*)d_in[2];   // (24,4,4)
  const float* transl = (const float*)d_in[3];   // (4,3)
  const float* yrp    = (const float*)d_in[4];   // (4,3)
  float* ws  = (float*)d_ws;
  float* out = (float*)d_out;                    // (4,64,200,200) f32

  float* pixX = ws + WS_PIXO;
  float* pixY = ws + WS_PIXO + NPIX;

  ipm_setup<<<1, 32, 0, stream>>>(Ks, RTs, transl, yrp, ws);

  // 2500 chunks of 16 points per view; 8 waves (chunks) per 256-thread block
  dim3 gProj((NPTS / 16 + 7) / 8, NK);
  ipm_project<<<gProj, 256, 0, stream>>>(ws, pixX, pixY);

  int nthreads = NB * NPTS;                      // 160000
  ipm_compose<<<(nthreads + 255) / 256, 256, 0, stream>>>(images, pixX, pixY, out);
}